// SimpleGNNDenoiser_36541581754987
// MI455X (gfx1250) — compile-verified
//
#include <hip/hip_runtime.h>

typedef __attribute__((ext_vector_type(2))) float v2f;
typedef __attribute__((ext_vector_type(8))) float v8f;

#define NODE_STRIDE 64  // padded feature stride (max hidden dim)

// ---------------- degree / normalization ----------------

__global__ void k_deg_init(float* __restrict__ deg, int n) {
    int i = blockIdx.x * blockDim.x + threadIdx.x;
    if (i < n) deg[i] = 1.0f;  // self-loop contribution
}

__global__ void k_deg_count(const int* __restrict__ dst, float* __restrict__ deg, int e) {
    int i = blockIdx.x * blockDim.x + threadIdx.x;
    if (i < e) atomicAdd(&deg[dst[i]], 1.0f);
}

__global__ void k_dinv(float* __restrict__ dinv, int n) {
    int i = blockIdx.x * blockDim.x + threadIdx.x;
    if (i < n) dinv[i] = rsqrtf(dinv[i]);  // deg >= 1 always (self loop)
}

// ---------------- input feature assembly ----------------

__global__ void k_build_x(const float* __restrict__ coords, const int* __restrict__ types,
                          const float* __restrict__ emb, float* __restrict__ x, int n) {
    int i = blockIdx.x * blockDim.x + threadIdx.x;
    if (i >= n) return;
    float* row = x + (size_t)i * NODE_STRIDE;
    row[0] = coords[3 * i + 0];
    row[1] = coords[3 * i + 1];
    row[2] = coords[3 * i + 2];
    int t = types[i];
    row[3] = emb[3 * t + 0];
    row[4] = emb[3 * t + 1];
    row[5] = emb[3 * t + 2];
#pragma unroll
    for (int c = 6; c < NODE_STRIDE; ++c) row[c] = 0.0f;
}

// ---------------- weight padding: each layer -> 64x64 zero-padded ----------------

__global__ void k_pad_w(const float* __restrict__ W1, const float* __restrict__ W2,
                        const float* __restrict__ W3, const float* __restrict__ W4,
                        const float* __restrict__ W5, float* __restrict__ wp) {
    int t = blockIdx.x * blockDim.x + threadIdx.x;  // 5*64*64 threads
    if (t >= 5 * 64 * 64) return;
    int l = t >> 12;
    int k = (t >> 6) & 63;
    int nn = t & 63;
    const int din[5]  = {6, 32, 64, 64, 32};
    const int dout[5] = {32, 64, 64, 32, 3};
    const float* W = (l == 0) ? W1 : (l == 1) ? W2 : (l == 2) ? W3 : (l == 3) ? W4 : W5;
    float v = 0.0f;
    if (k < din[l] && nn < dout[l]) v = W[k * dout[l] + nn];
    wp[t] = v;
}

// ---------------- dense transform H = X @ W via V_WMMA_F32_16X16X4_F32 ----------------
// One wave per 16-row strip. A: 16x4 f32 (2 VGPR/lane), B: 4x16 f32 (2 VGPR/lane),
// C/D: 16x16 f32 (8 VGPR/lane). Layouts per CDNA5 ISA 7.12.2.

__global__ void k_gemm(const float* __restrict__ X, const float* __restrict__ Wp,
                       float* __restrict__ H, int nrows, int kmax, int ntiles) {
    int wave = (blockIdx.x * blockDim.x + threadIdx.x) >> 5;
    int lane = threadIdx.x & 31;
    int row0 = wave * 16;
    if (row0 >= nrows) return;  // uniform per wave: EXEC stays all-1s for WMMA
    int half = lane >> 4;       // 0: lanes 0-15, 1: lanes 16-31
    int m = lane & 15;
    const float* xrow = X + (size_t)(row0 + m) * NODE_STRIDE;
    for (int nt = 0; nt < ntiles; ++nt) {
        int n0 = nt * 16;
        v8f c = {0.f, 0.f, 0.f, 0.f, 0.f, 0.f, 0.f, 0.f};
        for (int k = 0; k < kmax; k += 4) {
            v2f a, b;
            // A 16x4: VGPR0 = K = 0 (lanes 0-15) / K = 2 (lanes 16-31), VGPR1 = K = 1 / K = 3
            a.x = xrow[k + 2 * half + 0];
            a.y = xrow[k + 2 * half + 1];
            // B 4x16: N across lanes within a VGPR, same K split across lane halves
            b.x = Wp[(size_t)(k + 2 * half + 0) * NODE_STRIDE + n0 + m];
            b.y = Wp[(size_t)(k + 2 * half + 1) * NODE_STRIDE + n0 + m];
            c = __builtin_amdgcn_wmma_f32_16x16x4_f32(false, a, false, b, (short)0, c,
                                                      false, false);
        }
        // C/D: VGPR r, lanes 0-15 -> M=r, lanes 16-31 -> M=8+r
#pragma unroll
        for (int r = 0; r < 8; ++r) {
            H[(size_t)(row0 + half * 8 + r) * NODE_STRIDE + n0 + m] = c[r];
        }
    }
}

// ---------------- Y init: self-loop term + bias, zero padding columns ----------------

__global__ void k_init_y(const float* __restrict__ H, const float* __restrict__ dinv,
                         const float* __restrict__ bias, float* __restrict__ Y,
                         int n, int dout) {
    long long t = (long long)blockIdx.x * blockDim.x + threadIdx.x;
    if (t >= (long long)n * NODE_STRIDE) return;
    int i = (int)(t >> 6);
    int c = (int)(t & 63);
    float v = 0.0f;
    if (c < dout) {
        float d = dinv[i];
        v = H[t] * d * d + bias[c];
    }
    Y[t] = v;
}

// ---------------- edge scatter: Y[dst] += H[src] * dinv[src]*dinv[dst] ----------------
// One thread per (edge, 4-column chunk); f32 global atomics (L2-resident accumulators).

__global__ void k_scatter(const float* __restrict__ H, const float* __restrict__ dinv,
                          const int* __restrict__ src, const int* __restrict__ dstp,
                          float* __restrict__ Y, int e, int dout, int nchunks) {
    long long t = (long long)blockIdx.x * blockDim.x + threadIdx.x;
    long long total = (long long)e * nchunks;
    if (t >= total) return;
    int ch = (int)(t % nchunks);
    int ed = (int)(t / nchunks);
    // warm the edge-index stream (lowers to global_prefetch_b8)
    __builtin_prefetch(&src[ed + 1024], 0, 1);
    int s = src[ed];
    int d = dstp[ed];
    float norm = dinv[s] * dinv[d];
    int c0 = ch * 4;
    const float* hr = H + (size_t)s * NODE_STRIDE + c0;
    float* yr = Y + (size_t)d * NODE_STRIDE + c0;
#pragma unroll
    for (int j = 0; j < 4; ++j) {
        if (c0 + j < dout) atomicAdd(&yr[j], hr[j] * norm);
    }
}

__global__ void k_relu(float* __restrict__ Y, long long n) {
    long long t = (long long)blockIdx.x * blockDim.x + threadIdx.x;
    if (t < n) Y[t] = fmaxf(Y[t], 0.0f);
}

__global__ void k_out(const float* __restrict__ Y, float* __restrict__ out, int n) {
    int t = blockIdx.x * blockDim.x + threadIdx.x;
    if (t >= n * 3) return;
    int i = t / 3;
    int c = t % 3;
    out[t] = Y[(size_t)i * NODE_STRIDE + c];
}

// ---------------- host side ----------------

static inline size_t align256(size_t x) { return (x + 255) & ~(size_t)255; }

extern "C" void kernel_launch(void* const* d_in, const int* in_sizes, int n_in,
                              void* d_out, int out_size, void* d_ws, size_t ws_size,
                              hipStream_t stream) {
    const float* coords = (const float*)d_in[0];
    const int*   types  = (const int*)d_in[1];
    const int*   eidx   = (const int*)d_in[2];
    const float* emb    = (const float*)d_in[3];
    const float* W[5] = {(const float*)d_in[4], (const float*)d_in[6], (const float*)d_in[8],
                         (const float*)d_in[10], (const float*)d_in[12]};
    const float* B[5] = {(const float*)d_in[5], (const float*)d_in[7], (const float*)d_in[9],
                         (const float*)d_in[11], (const float*)d_in[13]};

    const int n = in_sizes[0] / 3;
    const int e = in_sizes[2] / 2;
    const int* src = eidx;
    const int* dst = eidx + e;

    // workspace carve-up
    char* ws = (char*)d_ws;
    size_t off = 0;
    float* dinv = (float*)(ws + off); off = align256(off + (size_t)n * sizeof(float));
    float* bufA = (float*)(ws + off); off = align256(off + (size_t)n * NODE_STRIDE * sizeof(float));
    float* bufB = (float*)(ws + off); off = align256(off + (size_t)n * NODE_STRIDE * sizeof(float));
    float* wp   = (float*)(ws + off); off = align256(off + (size_t)5 * 64 * 64 * sizeof(float));

    const int din_pad[5] = {8, 32, 64, 64, 32};
    const int dout[5]    = {32, 64, 64, 32, 3};

    const int TB = 256;
    // degrees -> dinv
    k_deg_init<<<(n + TB - 1) / TB, TB, 0, stream>>>(dinv, n);
    k_deg_count<<<(e + TB - 1) / TB, TB, 0, stream>>>(dst, dinv, e);
    k_dinv<<<(n + TB - 1) / TB, TB, 0, stream>>>(dinv, n);
    // features + padded weights
    k_build_x<<<(n + TB - 1) / TB, TB, 0, stream>>>(coords, types, emb, bufA, n);
    k_pad_w<<<(5 * 64 * 64 + TB - 1) / TB, TB, 0, stream>>>(W[0], W[1], W[2], W[3], W[4], wp);

    // GEMM grid: one wave per 16-row strip
    const int nWaves = (n + 15) / 16;
    const int gemmBlocks = (nWaves * 32 + TB - 1) / TB;
    const long long nFeat = (long long)n * NODE_STRIDE;
    const int featBlocks = (int)((nFeat + TB - 1) / TB);

    for (int l = 0; l < 5; ++l) {
        int ntiles = (dout[l] + 15) / 16;
        k_gemm<<<gemmBlocks, TB, 0, stream>>>(bufA, wp + (size_t)l * 64 * 64, bufB,
                                              n, din_pad[l], ntiles);
        k_init_y<<<featBlocks, TB, 0, stream>>>(bufB, dinv, B[l], bufA, n, dout[l]);
        int nchunks = (dout[l] + 3) / 4;
        long long total = (long long)e * nchunks;
        int scatBlocks = (int)((total + TB - 1) / TB);
        k_scatter<<<scatBlocks, TB, 0, stream>>>(bufB, dinv, src, dst, bufA, e, dout[l], nchunks);
        if (l < 4) k_relu<<<featBlocks, TB, 0, stream>>>(bufA, nFeat);
    }
    k_out<<<(n * 3 + TB - 1) / TB, TB, 0, stream>>>(bufA, (float*)d_out, n);
}